// MatchSAGE_68092411510982
// MI455X (gfx1250) — compile-verified
//
#include <hip/hip_runtime.h>
#include <hip/hip_bf16.h>

typedef float v2f __attribute__((ext_vector_type(2)));
typedef float v8f __attribute__((ext_vector_type(8)));

#define N_NODES 100000
#define E_G     640000
#define E_P     500000
#define E_NN    500000
#define D       128
#define HMDIM   30
#define ZSTRIDE 32
#define EPSB    1e-5f

// ---------------- utility: zero a float buffer ----------------
__global__ void zero_kernel(float* __restrict__ p, long n) {
    long i = (long)blockIdx.x * blockDim.x + threadIdx.x;
    if (i < n) p[i] = 0.0f;
}

// ---------------- degree: deg[dst] += 1 ----------------
__global__ void deg_kernel(const int* __restrict__ dst, float* __restrict__ deg, int ne) {
    int e = blockIdx.x * blockDim.x + threadIdx.x;
    if (e < ne) atomicAdd(&deg[dst[e]], 1.0f);
}

// ---------------- scatter: agg[dst,:] += x[src,:] (float4 per thread) ----------------
__global__ void scatter_kernel(const float* __restrict__ x, const int* __restrict__ src,
                               const int* __restrict__ dst, float* __restrict__ agg, int ne) {
    long t = (long)blockIdx.x * blockDim.x + threadIdx.x;
    int e = (int)(t >> 5);
    int c = (int)(t & 31) << 2;
    if (e >= ne) return;
    int s = src[e], d = dst[e];
    const float4 v = *reinterpret_cast<const float4*>(x + (long)s * D + c);
    float* a = agg + (long)d * D + c;
    atomicAdd(a + 0, v.x);
    atomicAdd(a + 1, v.y);
    atomicAdd(a + 2, v.z);
    atomicAdd(a + 3, v.w);
}

// ---------------- fused SAGE layer: out = act(X@Ws + (AGG/deg)@Wn + b) ----------------
// One wave computes a 16x16 tile with V_WMMA_F32_16X16X4_F32, K in steps of 4.
__global__ void sage_gemm_kernel(const float* __restrict__ X, const float* __restrict__ AGG,
                                 const float* __restrict__ deg, const float* __restrict__ Ws,
                                 const float* __restrict__ Wn, const float* __restrict__ bias,
                                 float* __restrict__ out, int nrows, int do_relu) {
    int wave = threadIdx.x >> 5;
    int lane = threadIdx.x & 31;
    int tm = (blockIdx.x * (blockDim.x >> 5) + wave) * 16;
    int tn = blockIdx.y * 16;
    if (tm >= nrows) return;                 // whole-wave uniform exit (EXEC stays all-1)
    int half = lane >> 4;                    // 0: K,K+1   1: K+2,K+3  (ISA 32-bit A/B layout)
    int l    = lane & 15;
    int row  = tm + l;
    float inv_deg = 1.0f / fmaxf(deg[row], 1.0f);
    const float* xr = X + (long)row * D;
    const float* ar = AGG + (long)row * D;

    v8f acc = {0.f,0.f,0.f,0.f,0.f,0.f,0.f,0.f};
    // X @ Ws
    for (int k = 0; k < D; k += 4) {
        int ka = k + (half << 1);
        v2f a, b;
        a.x = xr[ka];
        a.y = xr[ka + 1];
        b.x = Ws[ka * D + tn + l];
        b.y = Ws[(ka + 1) * D + tn + l];
        acc = __builtin_amdgcn_wmma_f32_16x16x4_f32(false, a, false, b, (short)0, acc, false, false);
    }
    // mean @ Wn
    for (int k = 0; k < D; k += 4) {
        int ka = k + (half << 1);
        v2f a, b;
        a.x = ar[ka] * inv_deg;
        a.y = ar[ka + 1] * inv_deg;
        b.x = Wn[ka * D + tn + l];
        b.y = Wn[(ka + 1) * D + tn + l];
        acc = __builtin_amdgcn_wmma_f32_16x16x4_f32(false, a, false, b, (short)0, acc, false, false);
    }
    // C layout: VGPR r -> lanes 0-15: M=r, lanes 16-31: M=r+8; N = lane&15
    float bj = bias[tn + l];
    #pragma unroll
    for (int r = 0; r < 8; r++) {
        int rr = tm + (half << 3) + r;
        float v = acc[r] + bj;
        if (do_relu) v = fmaxf(v, 0.0f);
        out[(long)rr * D + tn + l] = v;
    }
}

// ---------------- edge dot: out[e] = <hs[src[e]], hd[dst[e]]> ----------------
__global__ void edge_dot_kernel(const float* __restrict__ hs, const float* __restrict__ hd,
                                const int* __restrict__ src, const int* __restrict__ dst,
                                float* __restrict__ out, int ne) {
    int wave = threadIdx.x >> 5, lane = threadIdx.x & 31;
    int e = blockIdx.x * (blockDim.x >> 5) + wave;
    if (e >= ne) return;
    int s = src[e], d = dst[e];
    const float4 a = *reinterpret_cast<const float4*>(hs + (long)s * D + (lane << 2));
    const float4 b = *reinterpret_cast<const float4*>(hd + (long)d * D + (lane << 2));
    float v = a.x * b.x + a.y * b.y + a.z * b.z + a.w * b.w;
    #pragma unroll
    for (int off = 16; off > 0; off >>= 1) v += __shfl_down(v, off);
    if (lane == 0) out[e] = v;
}

// ---------------- edge MLP stage 1: z = (hs[gs]*hd[gd]) @ W1m + b1m, + BN sums ----------------
// Block = 256 threads handles 64 edges; products staged in LDS; 8 waves x WMMA tiles.
__global__ void mlp_z_kernel(const float* __restrict__ hs, const float* __restrict__ hd,
                             const int* __restrict__ src, const int* __restrict__ dst,
                             const float* __restrict__ W1m, const float* __restrict__ b1m,
                             float* __restrict__ z, float* __restrict__ bnsum,
                             float* __restrict__ bnsq, int ne) {
    __shared__ float P[64][132];             // stride 132 -> conflict-free fragment reads
    int tid = threadIdx.x;
    int e0 = blockIdx.x * 64;
    for (int idx = tid; idx < 64 * 32; idx += blockDim.x) {
        int i = idx >> 5;
        int c = (idx & 31) << 2;
        int e = e0 + i;
        if (e < ne) {
            int s = src[e], d = dst[e];
            float4 a = *reinterpret_cast<const float4*>(hs + (long)s * D + c);
            float4 b = *reinterpret_cast<const float4*>(hd + (long)d * D + c);
            P[i][c + 0] = a.x * b.x;
            P[i][c + 1] = a.y * b.y;
            P[i][c + 2] = a.z * b.z;
            P[i][c + 3] = a.w * b.w;
        } else {
            P[i][c + 0] = 0.f; P[i][c + 1] = 0.f; P[i][c + 2] = 0.f; P[i][c + 3] = 0.f;
        }
    }
    __syncthreads();

    int wave = tid >> 5, lane = tid & 31;
    int tm = (wave & 3) * 16;                // edge sub-tile within the 64
    int tn = (wave >> 2) * 16;               // output-column tile (0 or 16; HM=30 padded to 32)
    int half = lane >> 4, l = lane & 15;

    v8f acc = {0.f,0.f,0.f,0.f,0.f,0.f,0.f,0.f};
    for (int k = 0; k < D; k += 4) {
        int ka = k + (half << 1);
        v2f a, b;
        a.x = P[tm + l][ka];
        a.y = P[tm + l][ka + 1];
        int j = tn + l;
        b.x = (j < HMDIM) ? W1m[ka * HMDIM + j] : 0.0f;
        b.y = (j < HMDIM) ? W1m[(ka + 1) * HMDIM + j] : 0.0f;
        acc = __builtin_amdgcn_wmma_f32_16x16x4_f32(false, a, false, b, (short)0, acc, false, false);
    }

    int j = tn + l;
    if (j < HMDIM) {
        float bj = b1m[j];
        float s = 0.f, ss = 0.f;
        #pragma unroll
        for (int r = 0; r < 8; r++) {
            int e = e0 + tm + (half << 3) + r;
            if (e < ne) {
                float v = acc[r] + bj;
                z[(long)e * ZSTRIDE + j] = v;
                s += v;
                ss += v * v;
            }
        }
        atomicAdd(&bnsum[j], s);
        atomicAdd(&bnsq[j], ss);
    }
}

// ---------------- edge MLP stage 2: BN + relu + @W2m + b2m ----------------
__global__ void mlp_out_kernel(const float* __restrict__ z, const float* __restrict__ bnsum,
                               const float* __restrict__ bnsq, const float* __restrict__ gamma,
                               const float* __restrict__ beta, const float* __restrict__ W2m,
                               const float* __restrict__ b2m, float* __restrict__ out, int ne) {
    __shared__ float smu[HMDIM], sinv[HMDIM], sg[HMDIM], sb[HMDIM], sw[HMDIM];
    int tid = threadIdx.x;
    if (tid < HMDIM) {
        float inv_n = 1.0f / (float)ne;
        float mu = bnsum[tid] * inv_n;
        float var = bnsq[tid] * inv_n - mu * mu;
        smu[tid] = mu;
        sinv[tid] = rsqrtf(var + EPSB);
        sg[tid] = gamma[tid];
        sb[tid] = beta[tid];
        sw[tid] = W2m[tid];
    }
    __syncthreads();
    int e = blockIdx.x * blockDim.x + tid;
    if (e >= ne) return;
    const float* zr = z + (long)e * ZSTRIDE;
    float acc = b2m[0];
    #pragma unroll
    for (int j = 0; j < HMDIM; j++) {
        float zn = (zr[j] - smu[j]) * sinv[j] * sg[j] + sb[j];
        acc += fmaxf(zn, 0.0f) * sw[j];
    }
    out[e] = acc;
}

// ---------------- launcher ----------------
extern "C" void kernel_launch(void* const* d_in, const int* in_sizes, int n_in,
                              void* d_out, int out_size, void* d_ws, size_t ws_size,
                              hipStream_t stream) {
    const float* x1      = (const float*)d_in[0];
    const float* x2      = (const float*)d_in[1];
    const int*   g_src   = (const int*)d_in[2];
    const int*   g_dst   = (const int*)d_in[3];
    const int*   pos_src = (const int*)d_in[4];
    const int*   pos_dst = (const int*)d_in[5];
    const int*   neg_src = (const int*)d_in[6];
    const int*   neg_dst = (const int*)d_in[7];
    const float* Ws1     = (const float*)d_in[8];
    const float* Wn1     = (const float*)d_in[9];
    const float* b1      = (const float*)d_in[10];
    const float* Ws2     = (const float*)d_in[11];
    const float* Wn2     = (const float*)d_in[12];
    const float* b2      = (const float*)d_in[13];
    const float* W1m     = (const float*)d_in[14];
    const float* b1m     = (const float*)d_in[15];
    const float* gamma   = (const float*)d_in[16];
    const float* beta    = (const float*)d_in[17];
    const float* W2m     = (const float*)d_in[18];
    const float* b2m     = (const float*)d_in[19];
    float* out = (float*)d_out;
    float* ws  = (float*)d_ws;

    const long NM = (long)N_NODES * D;       // 12.8M floats
    float* deg  = ws;                        // [100096]  (padded N, 16B aligned)
    float* agg  = ws + 100096;               // [N*128]
    float* h1   = agg + NM;                  // [N*128]
    float* hsrc = h1 + NM;                   // [N*128]
    float* hdst = hsrc + NM;                 // [N*128]
    float* bn   = hdst + NM;                 // [64]: sum[0..30), sumsq[32..62)
    float* zbuf = agg;                       // reuse agg+h1 region: E*32 <= 2*N*128

    dim3 zgrid((unsigned)((NM + 255) / 256));
    dim3 gemm_grid((N_NODES / 16 + 7) / 8, D / 16);      // (782, 8), 8 waves/block
    int scat_grid = (int)(((long)E_G * 32 + 255) / 256); // 80000

    // degree (shared by all aggregations)
    zero_kernel<<<(100096 + 255) / 256, 256, 0, stream>>>(deg, 100096);
    deg_kernel<<<(E_G + 255) / 256, 256, 0, stream>>>(g_dst, deg, E_G);

    // ---- encode(x1) -> hsrc ----
    zero_kernel<<<zgrid, 256, 0, stream>>>(agg, NM);
    scatter_kernel<<<scat_grid, 256, 0, stream>>>(x1, g_src, g_dst, agg, E_G);
    sage_gemm_kernel<<<gemm_grid, 256, 0, stream>>>(x1, agg, deg, Ws1, Wn1, b1, h1, N_NODES, 1);
    zero_kernel<<<zgrid, 256, 0, stream>>>(agg, NM);
    scatter_kernel<<<scat_grid, 256, 0, stream>>>(h1, g_src, g_dst, agg, E_G);
    sage_gemm_kernel<<<gemm_grid, 256, 0, stream>>>(h1, agg, deg, Ws2, Wn2, b2, hsrc, N_NODES, 0);

    // ---- encode(x2) -> hdst ----
    zero_kernel<<<zgrid, 256, 0, stream>>>(agg, NM);
    scatter_kernel<<<scat_grid, 256, 0, stream>>>(x2, g_src, g_dst, agg, E_G);
    sage_gemm_kernel<<<gemm_grid, 256, 0, stream>>>(x2, agg, deg, Ws1, Wn1, b1, h1, N_NODES, 1);
    zero_kernel<<<zgrid, 256, 0, stream>>>(agg, NM);
    scatter_kernel<<<scat_grid, 256, 0, stream>>>(h1, g_src, g_dst, agg, E_G);
    sage_gemm_kernel<<<gemm_grid, 256, 0, stream>>>(h1, agg, deg, Ws2, Wn2, b2, hdst, N_NODES, 0);

    // ---- dot-product predictors ----
    edge_dot_kernel<<<(E_P + 7) / 8, 256, 0, stream>>>(hsrc, hdst, pos_src, pos_dst, out, E_P);
    edge_dot_kernel<<<(E_NN + 7) / 8, 256, 0, stream>>>(hsrc, hdst, neg_src, neg_dst, out + E_P, E_NN);

    // ---- MLP predictor over g edges (agg/h1 are dead now; zbuf aliases them) ----
    zero_kernel<<<1, 64, 0, stream>>>(bn, 64);
    mlp_z_kernel<<<(E_G + 63) / 64, 256, 0, stream>>>(hsrc, hdst, g_src, g_dst, W1m, b1m,
                                                      zbuf, bn, bn + 32, E_G);
    mlp_out_kernel<<<(E_G + 255) / 256, 256, 0, stream>>>(zbuf, bn, bn + 32, gamma, beta,
                                                          W2m, b2m, out + E_P + E_NN, E_G);
}